// MosaicMemory_3152505995411
// MI455X (gfx1250) — compile-verified
//
#include <hip/hip_runtime.h>
#include <hip/hip_bf16.h>

// MI455X / gfx1250, wave32. fp32 WMMA (V_WMMA_F32_16X16X4_F32) everywhere:
// the op is HBM-bound (64MB of u @ 23.3TB/s ~ 2.8us; only ~5.4 GFLOP), so
// fp32 matrix cores keep reference numerics (argmin/top-2 sensitive) at no
// bandwidth cost. Async global->LDS copies (ASYNCcnt) double-buffer the
// K-chunks so the HBM stream overlaps the WMMA work.

typedef __attribute__((ext_vector_type(2))) float v2f;
typedef __attribute__((ext_vector_type(8))) float v8f;

#define NTOK   16384   // B*T
#define DDIM   1024
#define NCOL   128     // 64 cols (read proj) + 64 cols (write proj)
#define KC_    256
#define NH_    32768   // NTOK * H

// ---- CDNA5 async global->LDS copy (guarded; sync fallback keeps it building)
#if defined(__has_builtin)
#if __has_builtin(__builtin_amdgcn_global_load_async_to_lds_b128) && \
    __has_builtin(__builtin_amdgcn_s_wait_asynccnt)
#define USE_ASYNC 1
#endif
#endif
#ifndef USE_ASYNC
#define USE_ASYNC 0
#endif

#if USE_ASYNC
typedef int v4i_t __attribute__((vector_size(16)));
typedef __attribute__((address_space(1))) v4i_t as1_v4i;   // global
typedef __attribute__((address_space(3))) v4i_t as3_v4i;   // LDS
#endif

__device__ __forceinline__ void copy_b128(void* lds_dst, const void* gsrc) {
#if USE_ASYNC
    // integer->pointer casts dodge the no-reinterpret-across-addrspace rule;
    // flat LDS addr bits [31:0] == LDS byte offset, flat global == as(1).
    __builtin_amdgcn_global_load_async_to_lds_b128(
        (as1_v4i*)(unsigned long long)gsrc,
        (as3_v4i*)(unsigned)(unsigned long long)lds_dst,
        0, 0);
#else
    *(float4*)lds_dst = *(const float4*)gsrc;
#endif
}
__device__ __forceinline__ void copies_done() {
#if USE_ASYNC
    __builtin_amdgcn_s_wait_asynccnt(0);   // LDS writes landed before barrier
#endif
}

// ---------------------------------------------------------------------------
// Kernel 1: Y[16384][128] = u @ [w_proj_r | w_proj_w]   (fp32 WMMA 16x16x4)
// 128 threads = 4 waves; block tile 64 rows x 128 cols; K double-buffered in
// chunks of 32 via async copies. Wave w: rows (w>>1)*32, cols (w&1)*64.
// ---------------------------------------------------------------------------
__global__ __launch_bounds__(128)
void mm_proj_kernel(const float* __restrict__ u,
                    const float* __restrict__ wr,
                    const float* __restrict__ ww,
                    float* __restrict__ Y) {
    __shared__ float ldsU[2][64][40];    // row stride 160B (16B aligned)
    __shared__ float ldsW[2][32][136];   // row stride 544B (16B aligned)

    const int tid  = threadIdx.x;
    const int lane = tid & 31;
    const int wave = tid >> 5;
    const int half = lane >> 4;          // K-half select (ISA 16x4 A/B layout)
    const int l16  = lane & 15;
    const int rg   = wave >> 1;          // 0..1
    const int cg   = wave & 1;           // 0..1
    const int r0   = blockIdx.x * 64;

    v8f acc[2][4];
#pragma unroll
    for (int rt = 0; rt < 2; ++rt)
#pragma unroll
        for (int ct = 0; ct < 4; ++ct) acc[rt][ct] = (v8f){};

    auto stage = [&](int buf, int kc) {
        // u chunk: 64 rows x 32 k = 512 float4, 4 per thread, coalesced
#pragma unroll
        for (int i = 0; i < 4; ++i) {
            int q = tid + i * 128, row = q >> 3, kq = (q & 7) * 4;
            copy_b128(&ldsU[buf][row][kq],
                      u + (size_t)(r0 + row) * DDIM + kc + kq);
        }
        // W chunk: 32 k x 128 n (cols 0..63 wr, 64..127 ww), 8 f4 per thread
#pragma unroll
        for (int i = 0; i < 8; ++i) {
            int q = tid + i * 128, k = q >> 5, nq = (q & 31) * 4;
            const float* src = (nq < 64) ? (wr + (size_t)(kc + k) * 64 + nq)
                                         : (ww + (size_t)(kc + k) * 64 + (nq - 64));
            copy_b128(&ldsW[buf][k][nq], src);
        }
    };

    stage(0, 0);
    copies_done();
    __syncthreads();

    for (int kci = 0; kci < DDIM / 32; ++kci) {
        const int cur = kci & 1;
        if (kci + 1 < DDIM / 32) stage(cur ^ 1, (kci + 1) * 32);  // prefetch

#pragma unroll
        for (int kk = 0; kk < 8; ++kk) {
            const int k = kk * 4;
            v2f a[2];
#pragma unroll
            for (int rt = 0; rt < 2; ++rt) {     // A: lanes0-15 K=k..k+1, 16-31 K=k+2..3
                int m = rg * 32 + rt * 16 + l16;
                a[rt].x = ldsU[cur][m][k + 2 * half + 0];
                a[rt].y = ldsU[cur][m][k + 2 * half + 1];
            }
            v2f b[4];
#pragma unroll
            for (int ct = 0; ct < 4; ++ct) {     // B mirror layout
                int n = cg * 64 + ct * 16 + l16;
                b[ct].x = ldsW[cur][k + 2 * half + 0][n];
                b[ct].y = ldsW[cur][k + 2 * half + 1][n];
            }
#pragma unroll
            for (int rt = 0; rt < 2; ++rt)
#pragma unroll
                for (int ct = 0; ct < 4; ++ct)
                    acc[rt][ct] = __builtin_amdgcn_wmma_f32_16x16x4_f32(
                        false, a[rt], false, b[ct], (short)0, acc[rt][ct], false, false);
        }
        copies_done();       // my async writes to the other buffer are in LDS
        __syncthreads();     // everyone done reading cur before it is re-staged
    }

    // C/D layout: VGPR j -> row j (lanes 0-15) / row j+8 (lanes 16-31), N=l16
#pragma unroll
    for (int rt = 0; rt < 2; ++rt)
#pragma unroll
        for (int ct = 0; ct < 4; ++ct)
#pragma unroll
            for (int j = 0; j < 8; ++j) {
                int row = r0 + rg * 32 + rt * 16 + j + 8 * half;
                int col = cg * 64 + ct * 16 + l16;
                Y[(size_t)row * NCOL + col] = acc[rt][ct][j];
            }
}

// ---------------------------------------------------------------------------
// Kernel 2: VQ routing. Grid (512, 2): x -> 32-token tile, y -> codebook r/w.
// Block 256 = 8 waves; wave = (token subtile 0/1) x (h,g in 0..3).
// s[16 tok][256 codes] = 2*(y @ cb^T) - e2 via 16 col-tiles x 4 WMMAs,
// running per-lane top-2, shfl_xor butterfly, LDS combine across g.
// ---------------------------------------------------------------------------
__global__ __launch_bounds__(256)
void route_kernel(const float* __restrict__ Y,
                  const float* __restrict__ cb_r,
                  const float* __restrict__ cb_w,
                  float* __restrict__ out) {
    __shared__ float ldsY[32][136];
    __shared__ float e2s[4][256];
    __shared__ int   s_i1[32][2][2];
    __shared__ int   s_i2[32][2][2];
    __shared__ float s_p1[32][2][2];
    __shared__ float s_p2[32][2][2];

    const int tid   = threadIdx.x;
    const int cbsel = blockIdx.y;
    const float* cb = cbsel ? cb_w : cb_r;

    // Stage this block's Y tile (32 tokens x 128 cols) coalesced via async
#pragma unroll
    for (int i = 0; i < 4; ++i) {
        int q = tid + i * 256, row = q >> 5, cq = (q & 31) * 4;
        copy_b128(&ldsY[row][cq],
                  Y + (size_t)(blockIdx.x * 32 + row) * NCOL + cq);
    }
    // e2[hg][code] = |codebook row|^2  (independent of the async copies)
#pragma unroll
    for (int hg = 0; hg < 4; ++hg) {
        const float* row = cb + ((size_t)hg * KC_ + tid) * 16;
        float s = 0.f;
#pragma unroll
        for (int d = 0; d < 16; ++d) s += row[d] * row[d];
        e2s[hg][tid] = s;
    }
    copies_done();
    __syncthreads();

    const int lane = tid & 31, wave = tid >> 5;
    const int half = lane >> 4, l16 = lane & 15;
    const int hg   = wave & 3;            // h = hg>>1, g = hg&1
    const int st   = wave >> 2;           // token subtile
    const int colbase = cbsel * 64 + hg * 16;

    // A frags (16 tokens x 16 dims) from LDS, reused across all 16 code tiles
    v2f av[4];
#pragma unroll
    for (int ks = 0; ks < 4; ++ks)
        av[ks] = *(const v2f*)&ldsY[st * 16 + l16][colbase + ks * 4 + 2 * half];

    float t1v[8], t2v[8];
    int   t1i[8], t2i[8];
#pragma unroll
    for (int j = 0; j < 8; ++j) { t1v[j] = -3.4e38f; t2v[j] = -3.4e38f; t1i[j] = 0; t2i[j] = 0; }

    const float* cbhg = cb + (size_t)hg * KC_ * 16;
    for (int ct = 0; ct < 16; ++ct) {
        const int code = ct * 16 + l16;
        v8f c = (v8f){};
        const float* crow = cbhg + (size_t)code * 16 + 2 * half;
#pragma unroll
        for (int ks = 0; ks < 4; ++ks) {
            v2f b; b.x = crow[ks * 4]; b.y = crow[ks * 4 + 1];
            c = __builtin_amdgcn_wmma_f32_16x16x4_f32(
                false, av[ks], false, b, (short)0, c, false, false);
        }
        const float e2v = e2s[hg][code];
#pragma unroll
        for (int j = 0; j < 8; ++j) {
            float v = 2.0f * c[j] - e2v;      // argmax == argmin dist (|y|^2 const)
            if (v > t1v[j])      { t2v[j] = t1v[j]; t2i[j] = t1i[j]; t1v[j] = v; t1i[j] = code; }
            else if (v > t2v[j]) { t2v[j] = v; t2i[j] = code; }
        }
    }

    // Cross-lane top-2 within each 16-lane half (rows j / j+8)
#pragma unroll
    for (int j = 0; j < 8; ++j) {
#pragma unroll
        for (int mask = 1; mask <= 8; mask <<= 1) {
            float u1 = __shfl_xor(t1v[j], mask, 32);
            int   k1 = __shfl_xor(t1i[j], mask, 32);
            float u2 = __shfl_xor(t2v[j], mask, 32);
            int   k2 = __shfl_xor(t2i[j], mask, 32);
            if (u1 > t1v[j]) { float tv = t1v[j]; int ti = t1i[j];
                               t1v[j] = u1; t1i[j] = k1; u1 = tv; k1 = ti; }
            if (u1 > t2v[j]) { t2v[j] = u1; t2i[j] = k1; }
            if (u2 > t2v[j]) { t2v[j] = u2; t2i[j] = k2; }
        }
    }

    const int h = hg >> 1, g = hg & 1;
    if (l16 == 0) {   // lane 0 -> rows 0..7, lane 16 -> rows 8..15
#pragma unroll
        for (int j = 0; j < 8; ++j) {
            int tl = st * 16 + j + 8 * half;
            float e  = __expf(t2v[j] - t1v[j]);   // softmax of 2 (shift cancels)
            float p1 = 1.0f / (1.0f + e);
            s_i1[tl][h][g] = t1i[j];
            s_i2[tl][h][g] = t2i[j];
            s_p1[tl][h][g] = p1;
            s_p2[tl][h][g] = e * p1;
        }
    }
    __syncthreads();

    if (tid < 64) {
        const int tl = tid >> 1, hh = tid & 1;
        const int n  = blockIdx.x * 32 + tl;
        int   c0[2]  = { s_i1[tl][hh][0], s_i2[tl][hh][0] };
        int   c1[2]  = { s_i1[tl][hh][1], s_i2[tl][hh][1] };
        float p0[2]  = { s_p1[tl][hh][0], s_p2[tl][hh][0] };
        float p1a[2] = { s_p1[tl][hh][1], s_p2[tl][hh][1] };
        // d_out (floats): idx_r[NH] idx_w[NH] cRi[4NH] cRw[4NH] cWi[4NH] cWw[4NH]
        const size_t base_idx = cbsel ? (size_t)NH_      : 0;
        const size_t base_ci  = cbsel ? (size_t)10 * NH_ : (size_t)2 * NH_;
        const size_t base_cw  = cbsel ? (size_t)14 * NH_ : (size_t)6 * NH_;
        const int p = n * 2 + hh;
        out[base_idx + p] = (float)(c0[0] + (c1[0] << 8));   // powK = [1,256]
#pragma unroll
        for (int cc = 0; cc < 4; ++cc) {                     // combos: (0,0)(0,1)(1,0)(1,1)
            int j0 = cc >> 1, j1 = cc & 1;
            out[base_ci + (size_t)p * 4 + cc] = (float)((c0[j0] + (c1[j1] << 8)) & 65535);
            out[base_cw + (size_t)p * 4 + cc] = p0[j0] * p1a[j1];
        }
    }
}

// ---------------------------------------------------------------------------
extern "C" void kernel_launch(void* const* d_in, const int* in_sizes, int n_in,
                              void* d_out, int out_size, void* d_ws, size_t ws_size,
                              hipStream_t stream) {
    const float* u   = (const float*)d_in[0];
    const float* wr  = (const float*)d_in[1];
    const float* ww  = (const float*)d_in[2];
    const float* cbr = (const float*)d_in[3];
    const float* cbw = (const float*)d_in[4];
    float* Y = (float*)d_ws;                 // 16384 x 128 f32 = 8 MB scratch

    mm_proj_kernel<<<dim3(NTOK / 64), dim3(128), 0, stream>>>(u, wr, ww, Y);
    route_kernel<<<dim3(NTOK / 32, 2), dim3(256), 0, stream>>>(Y, cbr, cbw, (float*)d_out);
}